// FixedKAN_60464549593381
// MI455X (gfx1250) — compile-verified
//
#include <hip/hip_runtime.h>
#include <hip/hip_bf16.h>

// ---------------------------------------------------------------------------
// FixedKAN forward, folded form:
//   out[b] = sum_i  P_i( tanh(x[b,i]) )
// where P_i is the degree-7 polynomial with monomial coefficients
//   a[i][m] = sum_k wsum[i*8+k] * C[k][m],   wsum = hweights^T @ coeffs
// C = Chebyshev T_k -> monomial conversion matrix (integers).
// ---------------------------------------------------------------------------

typedef __attribute__((ext_vector_type(2))) float v2f;
typedef __attribute__((ext_vector_type(4))) float v4f;
typedef __attribute__((ext_vector_type(8))) float v8f;

#define KAN_B   16384
#define KAN_IN  256
#define KAN_N   64
#define KAN_D   2048   // IN * 8

// ---------------------------------------------------------------------------
// Kernel 1: wsum[d] = sum_n coeffs[n][d] * h[n]  via V_WMMA_F32_16X16X4_F32.
// A tile (16x4 f32): every row r holds h[4*kb .. 4*kb+3]  (rows identical).
// B tile (4x16 f32): B[k][c] = coeffs[4*kb+k][d0+c].
// D rows are then all equal to the folded wsum chunk.
// Grid: 128 blocks x 32 threads (one wave = one 16-column tile of d).
// ---------------------------------------------------------------------------
__global__ void __launch_bounds__(32)
kan_fold_wmma(const float* __restrict__ coeffs,   // [64][2048]
              const float* __restrict__ h,        // [64]
              float* __restrict__ wsum)           // [2048]
{
    const int lane = threadIdx.x;          // 0..31, EXEC all ones for WMMA
    const int d0   = blockIdx.x * 16;
    const int n    = lane & 15;            // column within tile
    const int hi   = lane >> 4;            // 0: K rows {0,1}; 1: K rows {2,3}

    v8f acc = {};
    for (int kb = 0; kb < 16; ++kb) {      // 64 neurons / K=4
        const int k0 = 4 * kb + 2 * hi;
        v2f A, Bv;
        // A: 16x4, lane half selects K pair; rows replicated (same h per row)
        A.x = h[k0];
        A.y = h[k0 + 1];
        // B: 4x16, row K striped across lanes within a VGPR
        Bv.x = coeffs[(size_t)k0       * KAN_D + d0 + n];
        Bv.y = coeffs[(size_t)(k0 + 1) * KAN_D + d0 + n];
        acc = __builtin_amdgcn_wmma_f32_16x16x4_f32(
            /*neg_a=*/false, A, /*neg_b=*/false, Bv,
            /*c_mod=*/(short)0, acc, /*reuse_a=*/false, /*reuse_b=*/false);
    }
    // D row 0 (lanes 0-15, VGPR0) == wsum chunk (all rows equal)
    if (lane < 16) wsum[d0 + lane] = acc[0];
}

// ---------------------------------------------------------------------------
// Kernel 2: Chebyshev -> monomial per input dim (tiny; pure VALU).
// A is stored m-major: A[m*256 + i] so the main kernel's coefficient loads
// are coalesced.
// ---------------------------------------------------------------------------
__global__ void __launch_bounds__(256)
kan_convert(const float* __restrict__ w,   // wsum [256][8]
            float* __restrict__ A)         // [8][256] m-major
{
    const int i = threadIdx.x;             // 0..255 input dim
    const float w0 = w[i * 8 + 0], w1 = w[i * 8 + 1], w2 = w[i * 8 + 2],
                w3 = w[i * 8 + 3], w4 = w[i * 8 + 4], w5 = w[i * 8 + 5],
                w6 = w[i * 8 + 6], w7 = w[i * 8 + 7];
    A[0 * 256 + i] = w0 - w2 + w4 - w6;
    A[1 * 256 + i] = w1 - 3.f * w3 + 5.f * w5 - 7.f * w7;
    A[2 * 256 + i] = 2.f * w2 - 8.f * w4 + 18.f * w6;
    A[3 * 256 + i] = 4.f * w3 - 20.f * w5 + 56.f * w7;
    A[4 * 256 + i] = 8.f * w4 - 48.f * w6;
    A[5 * 256 + i] = 16.f * w5 - 112.f * w7;
    A[6 * 256 + i] = 32.f * w6;
    A[7 * 256 + i] = 64.f * w7;
}

// ---------------------------------------------------------------------------
// Fast tanh: gfx1250 has V_TANH_F32 (TRANS op).
// ---------------------------------------------------------------------------
__device__ __forceinline__ float kan_tanh(float v) {
#if __has_builtin(__builtin_amdgcn_tanhf)
    return __builtin_amdgcn_tanhf(v);
#else
    // tanh(v) = 1 - 2/(exp2(2*log2(e)*v)+1)
    const float e = __builtin_amdgcn_exp2f(v * 2.8853900817779268f);
    return 1.0f - 2.0f / (e + 1.0f);
#endif
}

// ---------------------------------------------------------------------------
// Kernel 3: main streaming kernel. One wave per row, 8 rows per wave.
// Lane l owns dims {4l..4l+3} and {128+4l..128+4l+3}: two coalesced 128-bit
// loads per row (non-temporal: x is single-use, 16 MB stream).
// Monomial coefficients held in 64 VGPRs per lane, loaded once per wave.
// ---------------------------------------------------------------------------
#define WAVES_PER_BLOCK 8
#define ROWS_PER_WAVE   8
// grid = 16384 / (8 rows * 8 waves) = 256 blocks

__global__ void __launch_bounds__(256)
kan_main(const float* __restrict__ x,    // [16384][256]
         const float* __restrict__ A,    // [8][256] m-major monomial coeffs
         float* __restrict__ out)        // [16384]
{
    const int lane = threadIdx.x & 31;
    const int wave = threadIdx.x >> 5;

    // dims handled by this lane
    int dims[8];
#pragma unroll
    for (int j = 0; j < 8; ++j)
        dims[j] = (j < 4) ? (4 * lane + j) : (128 + 4 * lane + (j - 4));

    // coefficient registers: c[j][m] = monomial coeff m of dim dims[j]
    float c[8][8];
#pragma unroll
    for (int m = 0; m < 8; ++m)
#pragma unroll
        for (int j = 0; j < 8; ++j)
            c[j][m] = A[m * 256 + dims[j]];

    const int gw   = blockIdx.x * WAVES_PER_BLOCK + wave;
    const int row0 = gw * ROWS_PER_WAVE;

    for (int r = 0; r < ROWS_PER_WAVE; ++r) {
        const int row = row0 + r;
        const v4f* xr = (const v4f*)(x + (size_t)row * KAN_IN);
        const v4f lo = __builtin_nontemporal_load(xr + lane);        // dims 4l..4l+3
        const v4f hi = __builtin_nontemporal_load(xr + 32 + lane);   // dims 128+4l..
        const float xv[8] = {lo.x, lo.y, lo.z, lo.w, hi.x, hi.y, hi.z, hi.w};

        float acc = 0.0f;
#pragma unroll
        for (int j = 0; j < 8; ++j) {
            const float t = kan_tanh(xv[j]);
            float p = c[j][7];
#pragma unroll
            for (int m = 6; m >= 0; --m)
                p = fmaf(p, t, c[j][m]);       // Horner: 7 FMAs
            acc += p;
        }

        // wave32 row reduction
#pragma unroll
        for (int off = 16; off > 0; off >>= 1)
            acc += __shfl_xor(acc, off, 32);
        if (lane == 0) out[row] = acc;
    }
}

// ---------------------------------------------------------------------------
// Launch: fold (WMMA) -> convert -> main, all on `stream` (ordered).
// Workspace: wsum (2048 f32) | A (2048 f32) = 16 KB of d_ws.
// ---------------------------------------------------------------------------
extern "C" void kernel_launch(void* const* d_in, const int* in_sizes, int n_in,
                              void* d_out, int out_size, void* d_ws, size_t ws_size,
                              hipStream_t stream) {
    const float* x      = (const float*)d_in[0];   // [16384*256]
    const float* coeffs = (const float*)d_in[1];   // [64*2048]
    const float* hw     = (const float*)d_in[2];   // [64]
    float* out  = (float*)d_out;                   // [16384]
    float* wsum = (float*)d_ws;                    // 2048 floats
    float* A    = wsum + KAN_D;                    // 2048 floats

    kan_fold_wmma<<<KAN_D / 16, 32, 0, stream>>>(coeffs, hw, wsum);
    kan_convert<<<1, 256, 0, stream>>>(wsum, A);
    kan_main<<<KAN_B / (WAVES_PER_BLOCK * ROWS_PER_WAVE), 32 * WAVES_PER_BLOCK, 0, stream>>>(x, A, out);
}